// MHC_76699525972619
// MI455X (gfx1250) — compile-verified
//
#include <hip/hip_runtime.h>
#include <math.h>

typedef __attribute__((ext_vector_type(2))) float v2f;
typedef __attribute__((ext_vector_type(8))) float v8f;

#define TILE_B   16
#define CHUNK    512
#define XS_STRIDE (CHUNK + 4)        // 516: WMMA A reads hit distinct banks
#define NWAVES   8
#define NC_DIM   8192
#define N_DIM    4
#define C_DIM    2048
#define NCHUNKS  (NC_DIM / CHUNK)    // 16
#define KSLICE   (CHUNK / NWAVES)    // 64 K per wave per chunk

// ---- pack kernel: phiT[32][8192] = [phi_res^T ; phi_pre^T ; phi_post^T ; 0] ----
__global__ __launch_bounds__(256) void phi_pack_kernel(
    const float* __restrict__ phi_res, const float* __restrict__ phi_pre,
    const float* __restrict__ phi_post, float* __restrict__ phiT)
{
    const int idx = blockIdx.x * 256 + threadIdx.x;   // 32*8192 total
    const int n = idx >> 13;          // 0..31
    const int k = idx & (NC_DIM - 1); // 0..8191
    float v;
    if (n < 16)       v = phi_res[(size_t)k * 16 + n];
    else if (n < 20)  v = phi_pre[(size_t)k * 4 + (n - 16)];
    else if (n < 24)  v = phi_post[(size_t)k * 4 + (n - 20)];
    else              v = 0.0f;
    phiT[(size_t)n * NC_DIM + k] = v;
}

__global__ __launch_bounds__(256) void mhc_fused_kernel(
    const float* __restrict__ x, const float* __restrict__ y,
    const float* __restrict__ phiT, const float* __restrict__ b_res,
    const float* __restrict__ b_pre, const float* __restrict__ b_post,
    const float* __restrict__ alpha_pre_p, const float* __restrict__ alpha_post_p,
    const float* __restrict__ alpha_res_p,
    float* __restrict__ z_out, float* __restrict__ out)
{
    __shared__ float xs[TILE_B * XS_STRIDE];   // 33 KB; reused as reduction buf
    __shared__ float S[TILE_B][25];            // reduced projections [row][col<24]
    __shared__ float ssqp[TILE_B][16];
    __shared__ float hpre[TILE_B][4];
    __shared__ float hpost[TILE_B][4];
    __shared__ float hres[TILE_B][16];

    const int tid  = threadIdx.x;
    const int lane = tid & 31;
    const int wave = tid >> 5;
    const int brow0 = blockIdx.x * TILE_B;

    // ---- pass 1: projection GEMM [16 x 8192] x [8192 x 32] via WMMA f32 ----
    const int lrow = tid >> 4;      // loader row 0..15
    const int lcol = tid & 15;      // loader sub-index 0..15
    const float* xrow = x + (size_t)(brow0 + lrow) * NC_DIM;
    float ssq = 0.0f;

    // WMMA lane mapping (A 16x4 f32 / B 4x16 f32)
    const int mlane = lane & 15;            // M (A) or N (B)
    const int koff  = (lane >> 4) * 2;      // lanes 16-31 supply K=2,3

    // per-lane B column base pointers (transposed, contiguous in K)
    const float* bp0 = phiT + (size_t)mlane * NC_DIM;          // res cols 0..15
    const float* bp1 = phiT + (size_t)(16 + mlane) * NC_DIM;   // pre/post/zero

    v8f acc0 = {0.f,0.f,0.f,0.f,0.f,0.f,0.f,0.f};
    v8f acc1 = {0.f,0.f,0.f,0.f,0.f,0.f,0.f,0.f};

    for (int ch = 0; ch < NCHUNKS; ++ch) {
        const int kbase = ch * CHUNK;
        // cooperative chunk load: 16 rows x 512 cols, coalesced float4
        #pragma unroll
        for (int j = 0; j < CHUNK / (16 * 4); ++j) {
            int c4 = lcol * 4 + j * 64;
            float4 v = *(const float4*)(xrow + kbase + c4);
            ssq += v.x * v.x + v.y * v.y + v.z * v.z + v.w * v.w;
            float* p = &xs[lrow * XS_STRIDE + c4];
            p[0] = v.x; p[1] = v.y; p[2] = v.z; p[3] = v.w;
        }
        if (ch + 1 < NCHUNKS)   // global_prefetch_b8 for next chunk (near)
            __builtin_prefetch(xrow + kbase + CHUNK + lcol * 4, 0, 3);
        __syncthreads();

        const int kk0 = wave * KSLICE + koff;   // this lane's first K pair
        #pragma unroll 4
        for (int s = 0; s < KSLICE / 4; ++s) {
            const int kk = kk0 + s * 4;
            v2f a   = *(const v2f*)(&xs[mlane * XS_STRIDE + kk]);   // ds_load_b64
            const int kg = kbase + kk;
            v2f bt0 = *(const v2f*)(bp0 + kg);                      // global b64
            v2f bt1 = *(const v2f*)(bp1 + kg);                      // global b64
            acc0 = __builtin_amdgcn_wmma_f32_16x16x4_f32(
                       false, a, false, bt0, (short)0, acc0, false, false);
            acc1 = __builtin_amdgcn_wmma_f32_16x16x4_f32(
                       false, a, false, bt1, (short)0, acc1, false, false);
        }
        __syncthreads();
    }

    // ---- cross-wave reduction of accumulators (reuse xs as scratch) ----
    ssqp[lrow][lcol] = ssq;
    float* red = xs;                              // 8 waves * 512 floats = 16 KB
    #pragma unroll
    for (int i = 0; i < 8; ++i) {
        red[wave * 512 +       i * 32 + lane] = acc0[i];
        red[wave * 512 + 256 + i * 32 + lane] = acc1[i];
    }
    __syncthreads();

    for (int p = tid; p < 512; p += 256) {
        float s = 0.0f;
        #pragma unroll
        for (int w = 0; w < NWAVES; ++w) s += red[w * 512 + p];
        const int t = p >> 8;                 // N-tile
        const int i = (p >> 5) & 7;           // C/D vgpr index
        const int l = p & 31;                 // lane
        const int m = i + ((l >= 16) ? 8 : 0);
        const int col = t * 16 + (l & 15);
        if (col < 24) S[m][col] = s;
    }
    __syncthreads();

    // ---- per-row scalars: r_inv, sigmoids, 4x4 Sinkhorn (16 threads) ----
    if (tid < TILE_B) {
        const int m = tid;
        float sq = 0.0f;
        #pragma unroll
        for (int u = 0; u < 16; ++u) sq += ssqp[m][u];
        const float ri = 90.50966799187809f / (sqrtf(sq) + 1e-6f); // sqrt(8192)/(||x||+eps)
        const float a_pre = alpha_pre_p[0], a_post = alpha_post_p[0], a_res = alpha_res_p[0];
        #pragma unroll
        for (int n = 0; n < 4; ++n) {
            float vp = a_pre  * (S[m][16 + n] * ri) + b_pre[n];
            float vq = a_post * (S[m][20 + n] * ri) + b_post[n];
            hpre[m][n]  = 1.0f / (1.0f + __expf(-vp));
            hpost[m][n] = 1.0f / (1.0f + __expf(-vq));
        }
        float Mm[16];
        #pragma unroll
        for (int e = 0; e < 16; ++e)
            Mm[e] = __expf(a_res * (S[m][e] * ri) + b_res[e]);
        for (int it = 0; it < 20; ++it) {
            #pragma unroll
            for (int n = 0; n < 4; ++n) {
                float inv = 1.0f / (Mm[n*4] + Mm[n*4+1] + Mm[n*4+2] + Mm[n*4+3] + 1e-12f);
                Mm[n*4] *= inv; Mm[n*4+1] *= inv; Mm[n*4+2] *= inv; Mm[n*4+3] *= inv;
            }
            #pragma unroll
            for (int j = 0; j < 4; ++j) {
                float inv = 1.0f / (Mm[j] + Mm[4+j] + Mm[8+j] + Mm[12+j] + 1e-12f);
                Mm[j] *= inv; Mm[4+j] *= inv; Mm[8+j] *= inv; Mm[12+j] *= inv;
            }
        }
        #pragma unroll
        for (int e = 0; e < 16; ++e) hres[m][e] = Mm[e];
    }
    __syncthreads();

    // ---- pass 2: mixing (streaming, float4 vectorized) ----
    for (int idx = tid; idx < TILE_B * (C_DIM / 4); idx += 256) {
        const int m  = idx >> 9;              // C_DIM/4 = 512 vectors per row
        const int c  = (idx & 511) * 4;
        const size_t b = (size_t)(brow0 + m);
        const size_t xb = b * NC_DIM + c;
        float4 x0 = *(const float4*)(x + xb);
        float4 x1 = *(const float4*)(x + xb + C_DIM);
        float4 x2 = *(const float4*)(x + xb + 2 * C_DIM);
        float4 x3 = *(const float4*)(x + xb + 3 * C_DIM);
        float4 yv = *(const float4*)(y + b * C_DIM + c);

        float hp0 = hpre[m][0], hp1 = hpre[m][1], hp2 = hpre[m][2], hp3 = hpre[m][3];
        float4 zv;
        zv.x = hp0*x0.x + hp1*x1.x + hp2*x2.x + hp3*x3.x;
        zv.y = hp0*x0.y + hp1*x1.y + hp2*x2.y + hp3*x3.y;
        zv.z = hp0*x0.z + hp1*x1.z + hp2*x2.z + hp3*x3.z;
        zv.w = hp0*x0.w + hp1*x1.w + hp2*x2.w + hp3*x3.w;
        *(float4*)(z_out + b * C_DIM + c) = zv;

        #pragma unroll
        for (int n = 0; n < 4; ++n) {
            float h0 = hres[m][n*4+0], h1 = hres[m][n*4+1];
            float h2 = hres[m][n*4+2], h3 = hres[m][n*4+3];
            float hq = hpost[m][n];
            float4 ov;
            ov.x = hq*yv.x + h0*x0.x + h1*x1.x + h2*x2.x + h3*x3.x;
            ov.y = hq*yv.y + h0*x0.y + h1*x1.y + h2*x2.y + h3*x3.y;
            ov.z = hq*yv.z + h0*x0.z + h1*x1.z + h2*x2.z + h3*x3.z;
            ov.w = hq*yv.w + h0*x0.w + h1*x1.w + h2*x2.w + h3*x3.w;
            *(float4*)(out + (b * N_DIM + n) * C_DIM + c) = ov;
        }
    }
}

extern "C" void kernel_launch(void* const* d_in, const int* in_sizes, int n_in,
                              void* d_out, int out_size, void* d_ws, size_t ws_size,
                              hipStream_t stream) {
    (void)in_sizes; (void)n_in; (void)ws_size; (void)out_size;
    const float* x        = (const float*)d_in[0];
    const float* y        = (const float*)d_in[1];
    const float* phi_res  = (const float*)d_in[2];
    const float* phi_pre  = (const float*)d_in[3];
    const float* phi_post = (const float*)d_in[4];
    const float* b_res    = (const float*)d_in[5];
    const float* b_pre    = (const float*)d_in[6];
    const float* b_post   = (const float*)d_in[7];
    const float* a_pre    = (const float*)d_in[8];
    const float* a_post   = (const float*)d_in[9];
    const float* a_res    = (const float*)d_in[10];

    float* phiT = (float*)d_ws;                               // 32*8192 floats = 1 MB
    float* z    = (float*)d_out;                              // [B, C]
    float* out  = (float*)d_out + (size_t)8192 * 2048;        // [B, N, C]

    phi_pack_kernel<<<dim3((32 * NC_DIM) / 256), 256, 0, stream>>>(
        phi_res, phi_pre, phi_post, phiT);

    mhc_fused_kernel<<<dim3(8192 / TILE_B), 256, 0, stream>>>(
        x, y, phiT, b_res, b_pre, b_post, a_pre, a_post, a_res, z, out);
}